// UltimateNoncommutativeKAOperator_38585986187666
// MI455X (gfx1250) — compile-verified
//
#include <hip/hip_runtime.h>
#include <hip/hip_bf16.h>
#include <math.h>

#define DIM 4096
#define ZETA2 1.6449340668482264  // pi^2/6
#define CUTOFF 1e-80

// First 30 primes (N_PRIME_CORR = 30); all <= DIM and < DIM-1, so P_DIAG == P_OFF == these.
__device__ __constant__ int kPrimes[30] = {
    2, 3, 5, 7, 11, 13, 17, 19, 23, 29,
    31, 37, 41, 43, 47, 53, 59, 61, 67, 71,
    73, 79, 83, 89, 97, 101, 103, 107, 109, 113};

typedef __attribute__((ext_vector_type(2))) double v2d;
typedef __attribute__((ext_vector_type(4))) int v4i;

#if __has_builtin(__builtin_amdgcn_global_store_async_from_lds_b128)
#define HAVE_ASYNC_ST 1
#else
#define HAVE_ASYNC_ST 0
#endif

// The async builtin only exists on the device (gfx1250) pass; the host pass
// legitimately lacks it. Assert only for the device pass.
#if defined(__HIP_DEVICE_COMPILE__)
static_assert(HAVE_ASYNC_ST,
              "gfx1250 async LDS->global store builtin not available on device pass");
#endif

// ---------------------------------------------------------------------------
// Kernel A: compute the real diagonal of the Hermitianized H, reduce for the
// Frobenius norm (off-diagonals are exactly zero after 0.5*(H+H^H)), and write
// diag[n] + reg into the workspace.
// ---------------------------------------------------------------------------
__global__ __launch_bounds__(1024) void diag_kernel(
    const double* __restrict__ s_real_p, const double* __restrict__ s_imag_p,
    const double* __restrict__ theta_p, double* __restrict__ diag_out) {
  __shared__ double red[1024];
  const int tid = threadIdx.x;

  const double sr = s_real_p[0];
  const double si = s_imag_p[0];
  const double th = theta_p[0];

  const double abss = sqrt(sr * sr + si * si);
  const double cstr = fmin(abss, 5.0);  // corr_strength
  const bool in_range = (fabs(sr) < 30.0) && (fabs(si) < 500.0);

  double vals[4];
  double acc = 0.0;
#pragma unroll
  for (int k = 0; k < 4; ++k) {
    const int n = tid + k * 1024 + 1;  // n in 1..4096
    const double logn = log((double)n);
    // Re(exp(-s*logn)) = exp(-sr*logn)*cos(si*logn)
    double re = exp(-sr * logn) * cos(si * logn);
    const bool safe = (-sr * logn) > -80.0;
    if (!(in_range || safe)) re = CUTOFF;
    // diagonal prime correction: theta*log(p)*corr_strength * (ZETA2/p), real.
    for (int i = 0; i < 30; ++i) {
      if (kPrimes[i] == n) re += th * logn * cstr * (ZETA2 / (double)n);
    }
    vals[k] = re;
    acc += re * re;  // |H_nn|^2 ; off-diagonals are exactly 0 so norm is diag-only
  }

  red[tid] = acc;
  __syncthreads();
  for (int s = 512; s > 0; s >>= 1) {
    if (tid < s) red[tid] += red[tid + s];
    __syncthreads();
  }
  if (tid == 0) red[0] = fmax(1e-22, sqrt(red[0]) * 1e-18);  // reg
  __syncthreads();
  const double reg = red[0];

#pragma unroll
  for (int k = 0; k < 4; ++k) {
    const int n = tid + k * 1024 + 1;
    diag_out[n - 1] = vals[k] + reg;
  }
}

// ---------------------------------------------------------------------------
// Kernel B: one block per row (4096 blocks x 256 threads). The entire 64 KB
// row of zeros is streamed with CDNA5 async LDS->global B128 stores (ASYNCcnt
// DMA path; zeros never touch VGPRs). After s_wait_asynccnt 0, the lane that
// owns the diagonal column overwrites it with (diag[row], 0) — same wave, so
// the wait orders the overwrite after the async zero to that address.
// Output is complex128 interleaved (re, im) doubles.
// ---------------------------------------------------------------------------
__global__ __launch_bounds__(256) void fill_kernel(
    const double* __restrict__ diag, double* __restrict__ out) {
  const int row = blockIdx.x;
  const int tid = threadIdx.x;
  double* rowp = out + (size_t)row * (2 * DIM);

#if HAVE_ASYNC_ST  // device pass (asserted above); host pass parses the #else
  __shared__ __align__(16) double zbuf[512];  // 16 zero bytes per lane
  zbuf[tid] = 0.0;
  zbuf[tid + 256] = 0.0;
  __syncthreads();
  __attribute__((address_space(3))) v4i* lsrc =
      (__attribute__((address_space(3))) v4i*)&zbuf[(size_t)tid * 2];

#pragma unroll
  for (int it = 0; it < 16; ++it) {
    const int col = it * 256 + tid;
    __builtin_amdgcn_global_store_async_from_lds_b128(
        (__attribute__((address_space(1))) v4i*)(rowp + (size_t)col * 2), lsrc,
        0, 0);
  }

#if __has_builtin(__builtin_amdgcn_s_wait_asynccnt)
  __builtin_amdgcn_s_wait_asynccnt(0);
#else
  asm volatile("s_wait_asynccnt 0" ::: "memory");
#endif

#else  // host-parse fallback: plain non-temporal B128 zero stores
  const v2d z = {0.0, 0.0};
#pragma unroll
  for (int it = 0; it < 16; ++it) {
    const int col = it * 256 + tid;
    __builtin_nontemporal_store(z, (v2d*)(rowp + (size_t)col * 2));
  }
#endif

  // The lane that issued the zero to the diagonal address rewrites it.
  if (tid == (row & 255)) {
    v2d val = {diag[row], 0.0};
    __builtin_nontemporal_store(val, (v2d*)(rowp + (size_t)row * 2));
  }
}

extern "C" void kernel_launch(void* const* d_in, const int* in_sizes, int n_in,
                              void* d_out, int out_size, void* d_ws,
                              size_t ws_size, hipStream_t stream) {
  const double* s_real = (const double*)d_in[0];
  const double* s_imag = (const double*)d_in[1];
  const double* theta = (const double*)d_in[2];
  double* out = (double*)d_out;  // complex128 interleaved (re, im)
  double* diag = (double*)d_ws;  // DIM doubles of scratch

  diag_kernel<<<1, 1024, 0, stream>>>(s_real, s_imag, theta, diag);
  fill_kernel<<<DIM, 256, 0, stream>>>(diag, out);
}